// Mask2FormerPanoptic_44770739093583
// MI455X (gfx1250) — compile-verified
//
#include <hip/hip_runtime.h>
#include <hip/hip_bf16.h>
#include <stdint.h>

// ---------------------------------------------------------------------------
// Mask2Former panoptic merge for MI455X (gfx1250).
// B=2, Q=100, C=133 (+1 null), masks 160x160 -> bilinear 4x -> 640x640.
// Output: [sem_seg 2*133*640*640][panoptic 2*640*640] as float32, concat.
// CDNA5 path: TDM tensor_load_to_lds stages 10x10 low-res tiles into LDS
// (16x reuse per texel), s_wait_tensorcnt + barrier before consumption.
// ---------------------------------------------------------------------------

#define BB      2
#define QQ      100
#define NC      133
#define NCP1    134
#define HM      160
#define WM      160
#define HO      640
#define WO      640
#define TILE    32              // output tile edge
#define LT      10              // low-res tile edge (8 + halo)
#define NTX     (WO / TILE)     // 20
#define NTY     (HO / TILE)     // 20

// workspace layout in 4-byte elements
#define OFF_SIG    0u                         // 2*100*160*160 = 5,120,000 f32
#define OFF_ESC    5120000u                   // 200 f32 (keep ? score : -1)
#define OFF_LAB    5120200u                   // 200 i32
#define OFF_KEEP   5120400u                   // 200 i32
#define OFF_MAREA  5120600u                   // 200 i32
#define OFF_OAREA  5120800u                   // 200 i32
#define OFF_ACC    5121000u                   // 200 i32
#define OFF_IDS    5121200u                   // 200 i32
#define OFF_CCNT   5121400u                   // 2*133 = 266 i32
#define OFF_CLIST  5121700u                   // 2*133*100 = 26,600 i32
#define OFF_MIDS   5148300u                   // 2*640*640 = 819,200 i32

typedef unsigned int u32x4 __attribute__((ext_vector_type(4)));
typedef int          i32x8 __attribute__((ext_vector_type(8)));
typedef int          i32x4 __attribute__((ext_vector_type(4)));

// --- Tensor Data Mover: DMA a 10x10 f32 tile (row stride 160 elems) from
// global memory into LDS at byte offset lds_off. D# per CDNA5 ISA 8.3/8.4.
// Toolchain here exposes the 6-arg builtin:
//   (uint32x4 g0, int32x8 g1, int32x4 g2, int32x4 g3, int32x8 pad, i32 cpol)
__device__ __forceinline__ void tdm_load_tile10x10(const float* gsrc,
                                                   unsigned lds_off) {
  unsigned long long ga = (unsigned long long)(uintptr_t)gsrc;
  u32x4 g0;
  g0[0] = 1u;                                   // count=1, user descriptor
  g0[1] = lds_off;                              // lds_addr (bytes)
  g0[2] = (unsigned)ga;                         // global_addr[31:0]
  g0[3] = ((unsigned)(ga >> 32) & 0x1FFFFFFu)   // global_addr[56:32]
          | (2u << 30);                         // type = 2 ("image")
  i32x8 g1;
  g1[0] = (int)(2u << 16);                      // data_size = 4 bytes
  g1[1] = (int)((unsigned)HM << 16);            // tensor_dim0[15:0]
  g1[2] = (int)((unsigned)HM << 16);            // tensor_dim1[15:0]
  g1[3] = (int)((unsigned)LT << 16);            // tile_dim0 = 10
  g1[4] = (int)LT;                              // tile_dim1 = 10, tile_dim2=0
  g1[5] = (int)WM;                              // tensor_dim0_stride = 160
  g1[6] = (int)((unsigned)(HM * WM) << 16);     // tensor_dim1_stride[15:0]
  g1[7] = 0;
  i32x4 z4 = {0, 0, 0, 0};
  i32x8 z8 = {0, 0, 0, 0, 0, 0, 0, 0};
  __builtin_amdgcn_tensor_load_to_lds(g0, g1, z4, z4, z8, 0);
}

__device__ __forceinline__ unsigned lds_byte_off(const void* p) {
  return (unsigned)(uintptr_t)p;   // addr[31:0] == LDS offset (flat aperture)
}

// bilinear 4x upsample, half-pixel centers: f = Y/4 - 0.375, edge clamp
__device__ __forceinline__ void axis_interp(int Y, int s, int& l0, int& l1,
                                            float& w) {
  float f  = (float)Y * 0.25f - 0.375f;
  float ff = floorf(f);
  int   i  = (int)ff;
  w = f - ff;
  int a = min(max(i, 0), HM - 1);
  int b = min(max(i + 1, 0), HM - 1);
  l0 = a - s;
  l1 = b - s;
}

__device__ __forceinline__ float interp_tile(const float* tq, int ly0, int ly1,
                                             int lx0, int lx1, float wy,
                                             float wx) {
  float t0 = tq[ly0 * LT + lx0] + wx * (tq[ly0 * LT + lx1] - tq[ly0 * LT + lx0]);
  float t1 = tq[ly1 * LT + lx0] + wx * (tq[ly1 * LT + lx1] - tq[ly1 * LT + lx0]);
  return t0 + wy * (t1 - t0);
}

// ---------------------------------------------------------------------------
// K1: per-(b,q) softmax stats. One wave per row of 134 logits.
__global__ void k_softmax(const float* __restrict__ logits,
                          float* __restrict__ effScore,
                          int* __restrict__ labels, int* __restrict__ keep) {
  const int row  = blockIdx.x;            // b*Q + q
  const int lane = threadIdx.x;           // 0..31
  const float* lg = logits + (size_t)row * NCP1;
  float m = -3.4e38f; int mi = 0;
  for (int c = lane; c < NCP1; c += 32) {
    float v = lg[c];
    if (v > m) { m = v; mi = c; }
  }
  for (int off = 16; off > 0; off >>= 1) {
    float om = __shfl_xor(m, off, 32);
    int   oi = __shfl_xor(mi, off, 32);
    if (om > m || (om == m && oi < mi)) { m = om; mi = oi; }
  }
  float s = 0.f;
  for (int c = lane; c < NCP1; c += 32) s += expf(lg[c] - m);
  for (int off = 16; off > 0; off >>= 1) s += __shfl_xor(s, off, 32);
  if (lane == 0) {
    float score = 1.0f / s;               // exp(lmax-lmax)/sum
    int kp = (mi < NC) && (score > 0.4f);
    effScore[row] = kp ? score : -1.0f;
    labels[row]   = mi;
    keep[row]     = kp;
  }
}

// K2: zero the per-query area counters.
__global__ void k_zero(int* __restrict__ marea, int* __restrict__ oarea) {
  int i = threadIdx.x;
  if (i < BB * QQ) { marea[i] = 0; oarea[i] = 0; }
}

// K3: sigmoid the low-res masks into workspace.
__global__ void k_sigmoid(const float* __restrict__ masks,
                          float* __restrict__ sig, int n) {
  int i = blockIdx.x * blockDim.x + threadIdx.x;
  if (i < n) sig[i] = 1.0f / (1.0f + expf(-masks[i]));
}

// K4: per 32x32 output tile: TDM-stage all 100 query tiles, per-pixel argmax,
// per-query areas (orig_area via ballot, mask_area via LDS counters).
__global__ void k_argmax(const float* __restrict__ sig,
                         const float* __restrict__ effScore,
                         int* __restrict__ maskIds,
                         int* __restrict__ marea, int* __restrict__ oarea) {
  __shared__ float tiles[QQ * LT * LT];   // 40 KB
  __shared__ float sks[QQ];
  __shared__ int   cntA[QQ], cntO[QQ];

  const int b  = blockIdx.z;
  const int tX = blockIdx.x, tY = blockIdx.y;
  const int t  = threadIdx.x;             // 0..255, 8 waves
  const int sX = min(max(tX * (TILE / 4) - 1, 0), WM - LT);
  const int sY = min(max(tY * (TILE / 4) - 1, 0), HM - LT);

  for (int i = t; i < QQ; i += 256) {
    sks[i]  = effScore[b * QQ + i];
    cntA[i] = 0;
    cntO[i] = 0;
  }
  const unsigned tbase = lds_byte_off(&tiles[0]);
  const int wave = t >> 5;
  for (int q = wave; q < QQ; q += 8) {
    const float* src = sig + ((size_t)(b * QQ + q) * (HM * WM) + sY * WM + sX);
    tdm_load_tile10x10(src, tbase + (unsigned)q * (LT * LT * 4));
  }
  __builtin_amdgcn_s_wait_tensorcnt(0);
  __syncthreads();

  for (int k = 0; k < 4; ++k) {
    int p  = t + k * 256;
    int py = p >> 5, px = p & 31;
    int Y = tY * TILE + py, X = tX * TILE + px;
    int ly0, ly1, lx0, lx1; float wy, wx;
    axis_interp(Y, sY, ly0, ly1, wy);
    axis_interp(X, sX, lx0, lx1, wx);
    float best = -2.0f; int bq = 0;
    for (int q = 0; q < QQ; ++q) {
      float v  = interp_tile(&tiles[q * LT * LT], ly0, ly1, lx0, lx1, wy, wx);
      unsigned long long bm = __ballot(v >= 0.5f);
      if ((t & 31) == 0) atomicAdd(&cntO[q], __popcll(bm));
      float sc = sks[q];
      float pm = (sc < 0.0f) ? -1.0f : sc * v;
      if (pm > best) { best = pm; bq = q; }   // ties -> first q, like argmax
    }
    atomicAdd(&cntA[bq], 1);
    maskIds[((size_t)b * HO + Y) * WO + X] = bq;
  }
  __syncthreads();
  for (int i = t; i < QQ; i += 256) {
    if (cntA[i]) atomicAdd(&marea[b * QQ + i], cntA[i]);
    if (cntO[i]) atomicAdd(&oarea[b * QQ + i], cntO[i]);
  }
}

// K5: acceptance test + cumsum ids + per-class accepted-query lists.
__global__ void k_accept(const int* __restrict__ keep,
                         const int* __restrict__ labels,
                         const int* __restrict__ marea,
                         const int* __restrict__ oarea,
                         int* __restrict__ acc, int* __restrict__ ids,
                         int* __restrict__ ccnt, int* __restrict__ clist) {
  if (threadIdx.x != 0 || blockIdx.x != 0) return;
  for (int i = 0; i < BB * NC; ++i) ccnt[i] = 0;
  for (int b = 0; b < BB; ++b) {
    int run = 0;
    for (int q = 0; q < QQ; ++q) {
      int i  = b * QQ + q;
      int ma = marea[i], oa = oarea[i];
      int a  = keep[i] && (ma > 0) && (oa > 0) &&
               ((float)ma > 0.8f * (float)oa);
      acc[i] = a;
      run += a;
      ids[i] = run;
      if (a) {
        int c = labels[i];
        int j = b * NC + c;
        clist[(size_t)j * QQ + ccnt[j]] = q;
        ccnt[j]++;
      }
    }
  }
}

// K6: panoptic id map (written as float into the concatenated output).
__global__ void k_panoptic(const int* __restrict__ maskIds,
                           const int* __restrict__ acc,
                           const int* __restrict__ ids,
                           float* __restrict__ outPan) {
  int i = blockIdx.x * blockDim.x + threadIdx.x;
  if (i >= BB * HO * WO) return;
  int b = i / (HO * WO);
  int q = maskIds[i];
  int r = b * QQ + q;
  outPan[i] = acc[r] ? (float)ids[r] : 0.0f;
}

// K7: semantic volume. One block per (b, class, 32x32 tile). Channels with no
// accepted query stream zeros; else TDM-stage <=8 query tiles per chunk and
// take the per-pixel max (atomic-free, deterministic).
__global__ void k_sem(const float* __restrict__ sig,
                      const int* __restrict__ ccnt,
                      const int* __restrict__ clist,
                      float* __restrict__ outSem) {
  __shared__ float tiles[8 * LT * LT];    // 3.2 KB
  const int bc = blockIdx.z;              // b*NC + c
  const int b  = bc / NC;
  const int tX = blockIdx.x, tY = blockIdx.y;
  const int t  = threadIdx.x;
  const int cnt = ccnt[bc];

  float accv[4] = {0.f, 0.f, 0.f, 0.f};
  if (cnt > 0) {
    const int sX = min(max(tX * (TILE / 4) - 1, 0), WM - LT);
    const int sY = min(max(tY * (TILE / 4) - 1, 0), HM - LT);
    const unsigned tbase = lds_byte_off(&tiles[0]);
    const int wave = t >> 5;
    for (int base = 0; base < cnt; base += 8) {
      int nq = min(8, cnt - base);
      if (wave < nq) {
        int q = clist[(size_t)bc * QQ + base + wave];
        const float* src =
            sig + ((size_t)(b * QQ + q) * (HM * WM) + sY * WM + sX);
        tdm_load_tile10x10(src, tbase + (unsigned)wave * (LT * LT * 4));
      }
      __builtin_amdgcn_s_wait_tensorcnt(0);
      __syncthreads();
      for (int k = 0; k < 4; ++k) {
        int p  = t + k * 256;
        int py = p >> 5, px = p & 31;
        int Y = tY * TILE + py, X = tX * TILE + px;
        int ly0, ly1, lx0, lx1; float wy, wx;
        axis_interp(Y, sY, ly0, ly1, wy);
        axis_interp(X, sX, lx0, lx1, wx);
        for (int j = 0; j < nq; ++j) {
          float v = interp_tile(&tiles[j * LT * LT], ly0, ly1, lx0, lx1, wy, wx);
          accv[k] = fmaxf(accv[k], v);
        }
      }
      __syncthreads();
    }
  }
  for (int k = 0; k < 4; ++k) {
    int p  = t + k * 256;
    int py = p >> 5, px = p & 31;
    int Y = tY * TILE + py, X = tX * TILE + px;
    outSem[((size_t)bc * HO + Y) * WO + X] = accv[k];
  }
}

// ---------------------------------------------------------------------------
extern "C" void kernel_launch(void* const* d_in, const int* in_sizes, int n_in,
                              void* d_out, int out_size, void* d_ws,
                              size_t ws_size, hipStream_t stream) {
  const float* logits = (const float*)d_in[0];   // [2,100,134]
  const float* masks  = (const float*)d_in[1];   // [2,100,160,160]
  (void)in_sizes; (void)n_in; (void)out_size; (void)ws_size;

  float* wsf = (float*)d_ws;
  int*   wsi = (int*)d_ws;
  float* sig      = wsf + OFF_SIG;
  float* effScore = wsf + OFF_ESC;
  int*   labels   = wsi + OFF_LAB;
  int*   keep     = wsi + OFF_KEEP;
  int*   marea    = wsi + OFF_MAREA;
  int*   oarea    = wsi + OFF_OAREA;
  int*   acc      = wsi + OFF_ACC;
  int*   ids      = wsi + OFF_IDS;
  int*   ccnt     = wsi + OFF_CCNT;
  int*   clist    = wsi + OFF_CLIST;
  int*   maskIds  = wsi + OFF_MIDS;

  float* outSem = (float*)d_out;                          // 2*133*640*640
  float* outPan = outSem + (size_t)BB * NC * HO * WO;     // 2*640*640

  k_softmax<<<BB * QQ, 32, 0, stream>>>(logits, effScore, labels, keep);
  k_zero<<<1, 256, 0, stream>>>(marea, oarea);
  {
    int n = BB * QQ * HM * WM;
    k_sigmoid<<<(n + 255) / 256, 256, 0, stream>>>(masks, sig, n);
  }
  k_argmax<<<dim3(NTX, NTY, BB), 256, 0, stream>>>(sig, effScore, maskIds,
                                                   marea, oarea);
  k_accept<<<1, 1, 0, stream>>>(keep, labels, marea, oarea, acc, ids, ccnt,
                                clist);
  {
    int n = BB * HO * WO;
    k_panoptic<<<(n + 255) / 256, 256, 0, stream>>>(maskIds, acc, ids, outPan);
  }
  k_sem<<<dim3(NTX, NTY, BB * NC), 256, 0, stream>>>(sig, ccnt, clist, outSem);
}